// GIN_Triplet_unit_21114059227217
// MI455X (gfx1250) — compile-verified
//
#include <hip/hip_runtime.h>

typedef __attribute__((ext_vector_type(2))) float v2f;
typedef __attribute__((ext_vector_type(8))) float v8f;

#define NNODES 50000
#define NEDGES 800000
#define HID    128
#define BN_EPS 1e-5f

// ---------------------------------------------------------------------------
// 1) agg = x (residual folded in); zero BN statistics accumulators
// ---------------------------------------------------------------------------
__global__ void init_kernel(const float* __restrict__ x,
                            float* __restrict__ agg,
                            float* __restrict__ stats) {
    size_t i = (size_t)blockIdx.x * blockDim.x + threadIdx.x;
    const size_t n4 = (size_t)NNODES * HID / 4;
    if (i < n4) {
        ((float4*)agg)[i] = ((const float4*)x)[i];
    }
    if (i < 2 * HID) stats[i] = 0.0f;
}

// ---------------------------------------------------------------------------
// 2) scatter-add: one wave (32 lanes) per edge, float4 per lane (128 ch)
//    agg[dst] += x[src]   (x and agg are L2-resident: 25.6 MB each)
// ---------------------------------------------------------------------------
__global__ void scatter_kernel(const float* __restrict__ x,
                               const int* __restrict__ ei,
                               float* __restrict__ agg) {
    int edge = blockIdx.x * 8 + (threadIdx.x >> 5);
    int lane = threadIdx.x & 31;
    if (edge >= NEDGES) return;
    int src = ei[edge];
    int dst = ei[NEDGES + edge];
    float4 v = *(const float4*)(x + (size_t)src * HID + lane * 4);
    float* p = agg + (size_t)dst * HID + lane * 4;
    atomicAdd(p + 0, v.x);
    atomicAdd(p + 1, v.y);
    atomicAdd(p + 2, v.z);
    atomicAdd(p + 3, v.w);
}

// ---------------------------------------------------------------------------
// 3/4) GEMM: out[N,128] = act(A[N,128] @ W[128,128] + bias)
//   f32 WMMA 16x16x4. One wave per 16-row strip, 8 column tiles in v8f accs.
//
//   LDS layout for W (pair-interleaved + parity swizzle):
//     pair p = k/2 owns a 256-float row; float index within row for column c:
//       idx = (c ^ ((p&1)<<4)) * 2 + (k&1)
//     -> B fragment {W[k][c], W[k+1][c]} is one aligned ds_load_b64, and the
//        parity XOR (bit 5 of idx) puts the two lane-halves (which always
//        read pair rows of opposite parity) on disjoint LDS bank halves.
//
//   A-frag (16x4 f32, ISA 7.12.2): lanes 0-15 -> {K=0,K=1} of row M=lane;
//                                  lanes 16-31 -> {K=2,K=3} of row M=lane-16.
//   C/D (16x16 f32): VGPR i -> row M=i (lanes 0-15) / M=i+8 (lanes 16-31).
// ---------------------------------------------------------------------------
__global__ void __launch_bounds__(256)
gemm_kernel(const float* __restrict__ A,
            const float* __restrict__ W,
            const float* __restrict__ bias,
            float* __restrict__ out,
            int relu) {
    __shared__ float ldsW[HID * HID];  // 64 KB, pair-interleaved layout

    // cooperative fill: global reads coalesced, LDS writes swizzled
    for (int i = threadIdx.x; i < HID * HID; i += 256) {
        int k    = i >> 7;                         // W row (K index)
        int c    = i & 127;                        // W col (N index)
        int pair = k >> 1;
        int cs   = c ^ ((pair & 1) << 4);          // parity swizzle
        ldsW[pair * 256 + cs * 2 + (k & 1)] = W[i];
    }
    __syncthreads();

    int wave = threadIdx.x >> 5;
    int lane = threadIdx.x & 31;
    int strip = blockIdx.x * 8 + wave;             // 16-row strip index
    if (strip * 16 >= NNODES) return;              // wave-uniform exit

    int row0   = strip * 16;
    int lane15 = lane & 15;
    int khalf  = (lane >> 4) << 1;                 // 0 (lanes 0-15) / 2 (16-31)

    const float* arow = A + (size_t)(row0 + lane15) * HID + khalf;

    v8f acc[8] = {};

    // software-pipelined A fragments (prefetch next k-step before consuming)
    v2f a_cur = *(const v2f*)(arow);
    for (int k0 = 0; k0 < HID; k0 += 4) {
        v2f a_next = (k0 + 4 < HID) ? *(const v2f*)(arow + k0 + 4) : a_cur;

        int pair = (k0 + khalf) >> 1;
        int par  = pair & 1;
        const float* wbase = ldsW + pair * 256 + lane15 * 2;
#pragma unroll
        for (int n = 0; n < 8; ++n) {
            v2f b = *(const v2f*)(wbase + ((n ^ par) << 5));  // ds_load_b64
            acc[n] = __builtin_amdgcn_wmma_f32_16x16x4_f32(
                /*neg_a=*/false, a_cur, /*neg_b=*/false, b,
                /*c_mod=*/(short)0, acc[n],
                /*reuse_a=*/false, /*reuse_b=*/false);
        }
        a_cur = a_next;
    }

    int rbase = row0 + ((lane >> 4) << 3);         // +8 for upper lane half
#pragma unroll
    for (int n = 0; n < 8; ++n) {
        int col  = n * 16 + lane15;
        float bv = bias[col];
#pragma unroll
        for (int i = 0; i < 8; ++i) {
            float v = acc[n][i] + bv;
            if (relu) v = fmaxf(v, 0.0f);
            out[(size_t)(rbase + i) * HID + col] = v;
        }
    }
}

// ---------------------------------------------------------------------------
// 5) per-channel sum / sum-of-squares over node dimension
// ---------------------------------------------------------------------------
__global__ void reduce_kernel(const float* __restrict__ h,
                              float* __restrict__ stats) {
    int ch = threadIdx.x;          // 0..127
    float s = 0.0f, s2 = 0.0f;
    for (int r = blockIdx.x; r < NNODES; r += gridDim.x) {
        float v = h[(size_t)r * HID + ch];
        s  += v;
        s2 += v * v;
    }
    atomicAdd(&stats[ch],       s);
    atomicAdd(&stats[HID + ch], s2);
}

// ---------------------------------------------------------------------------
// 6) batchnorm (training-mode batch stats), in place over d_out
// ---------------------------------------------------------------------------
__global__ void bn_kernel(float* __restrict__ h,
                          const float* __restrict__ stats,
                          const float* __restrict__ gamma,
                          const float* __restrict__ beta) {
    size_t i = (size_t)blockIdx.x * blockDim.x + threadIdx.x;
    if (i >= (size_t)NNODES * HID) return;
    int ch = (int)(i & (HID - 1));
    const float invN = 1.0f / (float)NNODES;
    float mean = stats[ch] * invN;
    float var  = stats[HID + ch] * invN - mean * mean;
    h[i] = (h[i] - mean) * rsqrtf(var + BN_EPS) * gamma[ch] + beta[ch];
}

// ---------------------------------------------------------------------------
extern "C" void kernel_launch(void* const* d_in, const int* in_sizes, int n_in,
                              void* d_out, int out_size, void* d_ws, size_t ws_size,
                              hipStream_t stream) {
    const float* x     = (const float*)d_in[0];
    const int*   ei    = (const int*)d_in[1];   // [2, NEDGES] int32
    const float* W1    = (const float*)d_in[2];
    const float* b1    = (const float*)d_in[3];
    const float* W2    = (const float*)d_in[4];
    const float* b2    = (const float*)d_in[5];
    const float* gamma = (const float*)d_in[6];
    const float* beta  = (const float*)d_in[7];
    float* out = (float*)d_out;

    // workspace layout: agg[N*H] | h1[N*H] | stats[256]
    float* agg   = (float*)d_ws;
    float* h1    = agg + (size_t)NNODES * HID;
    float* stats = h1  + (size_t)NNODES * HID;

    const size_t total = (size_t)NNODES * HID;          // 6.4M
    const int initBlocks = (int)((total / 4 + 255) / 256);

    init_kernel<<<initBlocks, 256, 0, stream>>>(x, agg, stats);

    scatter_kernel<<<(NEDGES + 7) / 8, 256, 0, stream>>>(x, ei, agg);

    const int strips = NNODES / 16;                     // 3125
    const int gemmBlocks = (strips + 7) / 8;            // 391

    gemm_kernel<<<gemmBlocks, 256, 0, stream>>>(agg, W1, b1, h1, /*relu=*/1);
    gemm_kernel<<<gemmBlocks, 256, 0, stream>>>(h1,  W2, b2, out, /*relu=*/0);

    reduce_kernel<<<128, 128, 0, stream>>>(out, stats);

    bn_kernel<<<(int)((total + 255) / 256), 256, 0, stream>>>(out, stats, gamma, beta);
}